// GCNClassifier_80642305949903
// MI455X (gfx1250) — compile-verified
//
#include <hip/hip_runtime.h>

// ---- problem constants (match reference) ----
#define IN_DIM   128
#define HID_DIM  128
#define OUT_DIM  64
#define NUM_GRAPHS 64

typedef float v2f __attribute__((ext_vector_type(2)));
typedef float v8f __attribute__((ext_vector_type(8)));

// ---------------- utility: fill buffer with a constant ----------------
__global__ void fill_kernel(float* __restrict__ p, float v, int n) {
    int i = blockIdx.x * blockDim.x + threadIdx.x;
    if (i < n) p[i] = v;
}

// ---------------- degree: deg[dst[e]] += 1 (deg pre-filled with 1.0 = self loop) ----
__global__ void degree_kernel(const int* __restrict__ dst, float* __restrict__ deg, int E) {
    int e = blockIdx.x * blockDim.x + threadIdx.x;
    if (e < E) atomicAdd(&deg[dst[e]], 1.0f);
}

__global__ void rsqrt_kernel(const float* __restrict__ deg, float* __restrict__ dinv, int n) {
    int i = blockIdx.x * blockDim.x + threadIdx.x;
    if (i < n) dinv[i] = rsqrtf(deg[i]);   // deg >= 1 always (self loop)
}

// ---------------- f32 WMMA GEMM: C[M,N] = A[M,K] @ B[K,N] ----------------
// One wave32 per 16x16 output tile; K consumed 4 at a time with
// V_WMMA_F32_16X16X4_F32 (exact f32, no precision loss vs reference).
// Layouts per CDNA5 ISA 7.12.2:
//   A 16x4 : lane%16 = M row; lane/16 selects K pair {k0,k0+1} vs {k0+2,k0+3}
//   B 4x16 : lane%16 = N col; same half-wave K-pair split
//   C 16x16: VGPR v -> row (half*8 + v), lane%16 -> col
__global__ void gemm_wmma_f32(const float* __restrict__ A,
                              const float* __restrict__ B,
                              float* __restrict__ C,
                              int M, int N, int K) {
    const int lane  = threadIdx.x;                 // 0..31
    const int wave  = threadIdx.y;                 // 0..7
    const int tilesN = N >> 4;
    const int totalTiles = (M >> 4) * tilesN;
    const int tile = blockIdx.x * blockDim.y + wave;
    if (tile >= totalTiles) return;                // uniform per wave: EXEC stays all-1s

    const int tm   = tile / tilesN;
    const int tn   = tile - tm * tilesN;
    const int row0 = tm << 4;
    const int col0 = tn << 4;
    const int half = lane >> 4;                    // 0 or 1
    const int l    = lane & 15;

    v8f acc = {};
    const float* Arow = A + (size_t)(row0 + l) * K;
    const float* Bcol = B + (size_t)col0 + l;

    for (int k0 = 0; k0 < K; k0 += 4) {
        const int ka = k0 + (half << 1);
        v2f a, b;
        a.x = Arow[ka];
        a.y = Arow[ka + 1];
        b.x = Bcol[(size_t)ka * N];
        b.y = Bcol[(size_t)(ka + 1) * N];
        acc = __builtin_amdgcn_wmma_f32_16x16x4_f32(
            /*neg_a=*/false, a, /*neg_b=*/false, b,
            /*c_mod=*/(short)0, acc, /*reuse_a=*/false, /*reuse_b=*/false);
    }

#pragma unroll
    for (int v = 0; v < 8; ++v) {
        C[(size_t)(row0 + (half << 3) + v) * N + col0 + l] = acc[v];
    }
}

// ---------------- edge scatter: agg[dst] += h[src] * dinv[src]*dinv[dst] ----------
// D/4 threads per edge, float4 gather + 4 float atomics. All traffic is
// L2-resident on MI455X (h + agg + indices << 192MB global L2).
__global__ void edge_aggregate(const float* __restrict__ h,
                               const float* __restrict__ dinv,
                               const int* __restrict__ src,
                               const int* __restrict__ dst,
                               float* __restrict__ agg,
                               int E, int D) {
    const int perEdge = D >> 2;
    int tid = blockIdx.x * blockDim.x + threadIdx.x;
    int e = tid / perEdge;
    if (e >= E) return;
    int c = (tid - e * perEdge) << 2;

    const int s = src[e];
    const int d = dst[e];
    const float norm = dinv[s] * dinv[d];

    const float4 hv = *(const float4*)(h + (size_t)s * D + c);
    float* out = agg + (size_t)d * D + c;
    atomicAdd(out + 0, hv.x * norm);
    atomicAdd(out + 1, hv.y * norm);
    atomicAdd(out + 2, hv.z * norm);
    atomicAdd(out + 3, hv.w * norm);
}

// ---------------- self-loop term + bias (+ optional ReLU), elementwise ----------
// out[i,d] = act(agg[i,d] + h[i,d]*dinv[i]^2 + bias[d])   (in-place on agg is OK)
__global__ void finish_layer(const float* __restrict__ agg,
                             const float* __restrict__ h,
                             const float* __restrict__ dinv,
                             const float* __restrict__ bias,
                             float* __restrict__ out,
                             int Nn, int D, int do_relu) {
    const int perRow = D >> 2;
    int tid = blockIdx.x * blockDim.x + threadIdx.x;
    int i = tid / perRow;
    if (i >= Nn) return;
    int c = (tid - i * perRow) << 2;

    const float di = dinv[i];
    const float w  = di * di;
    const float4 av = *(const float4*)(agg + (size_t)i * D + c);
    const float4 hv = *(const float4*)(h   + (size_t)i * D + c);
    const float4 bv = *(const float4*)(bias + c);
    float4 r;
    r.x = av.x + hv.x * w + bv.x;
    r.y = av.y + hv.y * w + bv.y;
    r.z = av.z + hv.z * w + bv.z;
    r.w = av.w + hv.w * w + bv.w;
    if (do_relu) {
        r.x = fmaxf(r.x, 0.0f); r.y = fmaxf(r.y, 0.0f);
        r.z = fmaxf(r.z, 0.0f); r.w = fmaxf(r.w, 0.0f);
    }
    *(float4*)(out + (size_t)i * D + c) = r;
}

// ---------------- global mean pool ----------------
__global__ void count_nodes(const int* __restrict__ batch, float* __restrict__ cnt, int Nn) {
    int i = blockIdx.x * blockDim.x + threadIdx.x;
    if (i < Nn) atomicAdd(&cnt[batch[i]], 1.0f);
}

__global__ void pool_sum(const float* __restrict__ h,
                         const int* __restrict__ batch,
                         float* __restrict__ sums, int Nn, int D) {
    const int perRow = D >> 2;
    int tid = blockIdx.x * blockDim.x + threadIdx.x;
    int i = tid / perRow;
    if (i >= Nn) return;
    int c = (tid - i * perRow) << 2;
    const int g = batch[i];
    const float4 hv = *(const float4*)(h + (size_t)i * D + c);
    float* out = sums + (size_t)g * D + c;
    atomicAdd(out + 0, hv.x);
    atomicAdd(out + 1, hv.y);
    atomicAdd(out + 2, hv.z);
    atomicAdd(out + 3, hv.w);
}

__global__ void pool_div(const float* __restrict__ sums,
                         const float* __restrict__ cnt,
                         float* __restrict__ out, int G, int D) {
    int tid = blockIdx.x * blockDim.x + threadIdx.x;
    if (tid >= G * D) return;
    int g = tid / D;
    out[tid] = sums[tid] / fmaxf(cnt[g], 1.0f);
}

// ---------------- launcher ----------------
extern "C" void kernel_launch(void* const* d_in, const int* in_sizes, int n_in,
                              void* d_out, int out_size, void* d_ws, size_t ws_size,
                              hipStream_t stream) {
    const float* x    = (const float*)d_in[0];
    const int*   eidx = (const int*)  d_in[1];   // [2,E] flat: row0=src, row1=dst
    const int*   batch= (const int*)  d_in[2];
    const float* W1   = (const float*)d_in[3];
    const float* b1   = (const float*)d_in[4];
    const float* W2   = (const float*)d_in[5];
    const float* b2   = (const float*)d_in[6];
    float* out = (float*)d_out;

    const int Nn = in_sizes[0] / IN_DIM;   // 50000 (== 3125*16, tail-free)
    const int E  = in_sizes[1] / 2;        // 600000
    const int* src = eidx;
    const int* dst = eidx + E;

    // workspace layout (floats)
    float* ws   = (float*)d_ws;
    float* h1   = ws;                               // [Nn, HID]
    float* agg1 = h1   + (size_t)Nn * HID_DIM;      // [Nn, HID] -> becomes a1
    float* h2   = agg1 + (size_t)Nn * HID_DIM;      // [Nn, OUT]
    float* agg2 = h2   + (size_t)Nn * OUT_DIM;      // [Nn, OUT]
    float* deg  = agg2 + (size_t)Nn * OUT_DIM;      // [Nn]
    float* dinv = deg  + Nn;                        // [Nn]
    float* sums = dinv + Nn;                        // [G, OUT]
    float* cnts = sums + NUM_GRAPHS * OUT_DIM;      // [G]

    const int B = 256;
    auto blocks = [](long long n, int b) { return (unsigned)((n + b - 1) / b); };

    // init
    fill_kernel<<<blocks((long long)Nn * HID_DIM, B), B, 0, stream>>>(agg1, 0.0f, Nn * HID_DIM);
    fill_kernel<<<blocks((long long)Nn * OUT_DIM, B), B, 0, stream>>>(agg2, 0.0f, Nn * OUT_DIM);
    fill_kernel<<<blocks(Nn, B), B, 0, stream>>>(deg, 1.0f, Nn);             // self loops
    fill_kernel<<<blocks(NUM_GRAPHS * OUT_DIM, B), B, 0, stream>>>(sums, 0.0f, NUM_GRAPHS * OUT_DIM);
    fill_kernel<<<blocks(NUM_GRAPHS, B), B, 0, stream>>>(cnts, 0.0f, NUM_GRAPHS);

    // degrees + normalization
    degree_kernel<<<blocks(E, B), B, 0, stream>>>(dst, deg, E);
    rsqrt_kernel<<<blocks(Nn, B), B, 0, stream>>>(deg, dinv, Nn);

    // layer 1: h1 = x @ W1 (WMMA f32), scatter, bias+selfloop+relu
    {
        int tiles = (Nn >> 4) * (HID_DIM >> 4);
        dim3 blk(32, 8);
        gemm_wmma_f32<<<blocks(tiles, 8), blk, 0, stream>>>(x, W1, h1, Nn, HID_DIM, IN_DIM);
    }
    edge_aggregate<<<blocks((long long)E * (HID_DIM >> 2), B), B, 0, stream>>>(
        h1, dinv, src, dst, agg1, E, HID_DIM);
    finish_layer<<<blocks((long long)Nn * (HID_DIM >> 2), B), B, 0, stream>>>(
        agg1, h1, dinv, b1, agg1, Nn, HID_DIM, 1);

    // layer 2: h2 = a1 @ W2 (WMMA f32), scatter, bias+selfloop (no relu)
    {
        int tiles = (Nn >> 4) * (OUT_DIM >> 4);
        dim3 blk(32, 8);
        gemm_wmma_f32<<<blocks(tiles, 8), blk, 0, stream>>>(agg1, W2, h2, Nn, OUT_DIM, HID_DIM);
    }
    edge_aggregate<<<blocks((long long)E * (OUT_DIM >> 2), B), B, 0, stream>>>(
        h2, dinv, src, dst, agg2, E, OUT_DIM);
    finish_layer<<<blocks((long long)Nn * (OUT_DIM >> 2), B), B, 0, stream>>>(
        agg2, h2, dinv, b2, agg2, Nn, OUT_DIM, 0);

    // mean pool over graphs
    count_nodes<<<blocks(Nn, B), B, 0, stream>>>(batch, cnts, Nn);
    pool_sum<<<blocks((long long)Nn * (OUT_DIM >> 2), B), B, 0, stream>>>(
        agg2, batch, sums, Nn, OUT_DIM);
    pool_div<<<blocks(NUM_GRAPHS * OUT_DIM, B), B, 0, stream>>>(
        sums, cnts, out, NUM_GRAPHS, OUT_DIM);
}